// MobiusCollapseLayer_69741678952853
// MI455X (gfx1250) — compile-verified
//
#include <hip/hip_runtime.h>
#include <hip/hip_bf16.h>
#include <math.h>

// ---------------------------------------------------------------------------
// MobiusCollapseLayer for MI455X (gfx1250, wave32)
//   - fp32 WMMA 16x16x4 for all GEMMs (exact fp32 like the reference)
//   - A-block staging via TDM (tensor_load_to_lds + s_wait_tensorcnt) when the
//     builtin exists; async global->LDS b32 ops (ASYNCcnt) otherwise/elsewhere
//   - hardware pad (TDM pad_interval=256dw, pad_amount=1dw) == manual skew:
//     LDS row stride 1028 floats = 4 mod 64 banks -> conflict-free a-frags
// ---------------------------------------------------------------------------

typedef __attribute__((ext_vector_type(2))) float v2f;
typedef __attribute__((ext_vector_type(8))) float v8f;
typedef unsigned int u32x4 __attribute__((ext_vector_type(4)));
typedef int i32x4 __attribute__((ext_vector_type(4)));
typedef int i32x8 __attribute__((ext_vector_type(8)));

constexpr int Bb = 4, Ss = 2048, Dd = 1024, Pp = 9, Mm = 256;
constexpr int BS = Bb * Ss;            // 8192 rows
constexpr float EPSv = 1e-6f;
constexpr float TAUv = 1.0f;

constexpr int APAD = Dd + 4;   // 1028 floats: stride % 64 banks == 4
constexpr int HPAD = Mm + 4;   // 260 floats:  stride % 64 banks == 4

#if defined(__has_builtin)
#if __has_builtin(__builtin_amdgcn_tensor_load_to_lds)
#define USE_TDM 1
#else
#define USE_TDM 0
#endif
#else
#define USE_TDM 0
#endif

// with TDM padding, 1 pad dword is inserted after every 256 stream dwords,
// so float f of a row lives at (f + (f>>8)); without TDM we pack rows plain.
constexpr int kSkew = USE_TDM;

__device__ __forceinline__ int laneid() { return threadIdx.x & 31; }

// ---- WMMA f32 16x16x4 (codegen-confirmed) ---------------------------------
__device__ __forceinline__ v8f wmma4(v2f a, v2f b, v8f c) {
  return __builtin_amdgcn_wmma_f32_16x16x4_f32(false, a, false, b, (short)0, c,
                                               false, false);
}

// ---- async global -> LDS (ASYNCcnt; asm-confirmed round 2) ----------------
__device__ __forceinline__ void async_ld_b32(unsigned lds_byte_off,
                                             const void* g) {
  asm volatile("global_load_async_to_lds_b32 %0, %1, off"
               :: "v"(lds_byte_off), "v"(g)
               : "memory");
}
__device__ __forceinline__ void wait_async0() {
  asm volatile("s_wait_asynccnt 0" ::: "memory");
}

// Stage 16 rows x Dd floats into LDS (row stride APAD, optional TDM skew).
// 256 threads; each moves 4 floats per row with async b32 ops.
__device__ __forceinline__ void stage16_async(float* dst_s,
                                              const float* __restrict__ src) {
  unsigned base = (unsigned)(uintptr_t)dst_s;  // low 32 bits == LDS offset
  int t = threadIdx.x;
  int f = t * 4;
  unsigned lo = (unsigned)(f + (kSkew ? (f >> 8) : 0));
#pragma unroll
  for (int r = 0; r < 16; ++r) {
    unsigned off = base + ((unsigned)(r * APAD) + lo) * 4u;
    const float* s = src + (size_t)r * Dd + f;
#pragma unroll
    for (int j = 0; j < 4; ++j) async_ld_b32(off + 4u * j, s + j);
  }
}

#if USE_TDM
// One TDM op moves the whole 16x1024 f32 tile; hardware inserts the LDS pad.
__device__ __forceinline__ void tdm_load16(float* dst_s,
                                           const float* __restrict__ src) {
  unsigned long long ga = (unsigned long long)(uintptr_t)src;
  u32x4 g0;
  g0[0] = 1u;                                   // count=1 (valid descriptor)
  g0[1] = (unsigned)(uintptr_t)dst_s;           // lds_addr (bytes)
  g0[2] = (unsigned)ga;                         // global_addr[31:0]
  g0[3] = (unsigned)((ga >> 32) & 0x01FFFFFFu)  // global_addr[56:32]
          | (2u << 30);                         // type = 2 ("image")
  i32x8 g1;
  g1[0] = (int)((2u << 16)      // data_size = 4 bytes
                | (1u << 20)    // pad_enable
                | (7u << 22));  // pad_interval = 256 dwords (pad_amount=1 dw)
  g1[1] = (int)(((unsigned)Dd & 0xFFFFu) << 16);        // tensor_dim0 lo16
  g1[2] = (int)(((unsigned)Dd >> 16) | (16u << 16));    // dim0 hi | dim1 lo
  g1[3] = (int)(((unsigned)Dd & 0xFFFFu) << 16);        // dim1 hi=0 | tile_dim0
  g1[4] = 16;                                           // tile_dim1 (rows)
  g1[5] = Dd;                                           // tensor_dim0_stride
  g1[6] = 0;
  g1[7] = 0;
  i32x4 z4 = {0, 0, 0, 0};
#if defined(__clang_major__) && __clang_major__ >= 23
  i32x8 z8 = {0, 0, 0, 0, 0, 0, 0, 0};
  __builtin_amdgcn_tensor_load_to_lds(g0, g1, z4, z4, z8, 0);
#else
  __builtin_amdgcn_tensor_load_to_lds(g0, g1, z4, z4, 0);
#endif
}
#endif

// stage A block with TDM if available (wave 0 issues; caller barriers after)
__device__ __forceinline__ void stageA(float* dst_s,
                                       const float* __restrict__ src) {
#if USE_TDM
  if (threadIdx.x < 32) {
    tdm_load16(dst_s, src);
    __builtin_amdgcn_s_wait_tensorcnt(0);
  }
#else
  stage16_async(dst_s, src);
  wait_async0();
#endif
}

// A fragment from a staged (padded/skewed) LDS A block.
__device__ __forceinline__ v2f a_fragA(const float* As, int k) {
  int l = laneid();
  int f = k + ((l >> 4) << 1);
  const float* p =
      As + (size_t)(l & 15) * APAD + f + (kSkew ? (f >> 8) : 0);
  v2f r; r.x = p[0]; r.y = p[1];
  return r;
}

// generic A fragment (row-major, stride lda, no skew) — used for h_s
__device__ __forceinline__ v2f a_frag(const float* Arow0, int lda, int k) {
  int l = laneid();
  const float* p = Arow0 + (size_t)(l & 15) * lda + k + ((l >> 4) << 1);
  v2f r; r.x = p[0]; r.y = p[1];
  return r;
}

// B fragment: 4x16 (KxN) f32; VGPR r covers rows {r, r+2} across lane halves.
__device__ __forceinline__ v2f b_frag(const float* Bcol0, int ldb, int k) {
  int l = laneid();
  const float* p = Bcol0 + (size_t)(k + ((l >> 4) << 1)) * ldb + (l & 15);
  v2f r; r.x = p[0]; r.y = p[ldb];
  return r;
}

// C/D 16x16 tile: element (r, lane): row = r + 8*(lane>=16), col = lane&15.
__device__ __forceinline__ void store_tile(float* C, int ldc, const v8f& c) {
  int l = laneid();
  int col = l & 15, rhi = (l >> 4) << 3;
#pragma unroll
  for (int r = 0; r < 8; ++r) C[(size_t)(r + rhi) * ldc + col] = c[r];
}

// learned rotation of output columns 0,1 (call only on the tile with col0==0)
__device__ __forceinline__ void rot_tile_cols01(v8f& c, float cth, float sth) {
  int l = laneid();
  bool in01 = (l & 15) < 2;
#pragma unroll
  for (int r = 0; r < 8; ++r) {
    float v = c[r];
    float t = __shfl_xor(v, 1, 32);
    float rv = (l & 1) ? (t * sth + v * cth) : (v * cth - t * sth);
    c[r] = in01 ? rv : v;
  }
}

// ---------------------------------------------------------------------------
// K1: neighbor mixer  x_mixed = w0*x + w1*x[s-1] + w2*x[s+1]
// ---------------------------------------------------------------------------
__global__ void k_mix(const float* __restrict__ x, const float* __restrict__ w,
                      float* __restrict__ xm) {
  size_t i = (size_t)blockIdx.x * 256 + threadIdx.x;
  int s = (int)((i / Dd) % Ss);
  float v = x[i] * w[0];
  if (s > 0) v += x[i - Dd] * w[1];
  if (s < Ss - 1) v += x[i + Dd] * w[2];
  xm[i] = v;
}

// ---------------------------------------------------------------------------
// K2: observer hidden  h = tanh(rot(concat(xm,col) @ W1 + b1, th1))  [BS, M]
// A halves staged via async b32 ops; wave w owns col tiles {2w, 2w+1}.
// ---------------------------------------------------------------------------
__global__ __launch_bounds__(256) void k_obs_h(
    const float* __restrict__ xm, const float* __restrict__ col,
    const float* __restrict__ W1, const float* __restrict__ b1,
    const float* __restrict__ th1, float* __restrict__ h) {
  __shared__ __align__(16) float a_s[16 * APAD];  // 65.8 KB
  int tm = blockIdx.x;
  int w = threadIdx.x >> 5;
  int l = laneid();
  v8f c0 = {}, c1 = {};
  const float* Bc0 = W1 + (2 * w) * 16;
  const float* Bc1 = W1 + (2 * w + 1) * 16;
  for (int half = 0; half < 2; ++half) {
    stage16_async(a_s, (half ? col : xm) + (size_t)tm * 16 * Dd);
    wait_async0();
    __syncthreads();
    for (int k = 0; k < Dd; k += 4) {
      v2f a = a_fragA(a_s, k);
      c0 = wmma4(a, b_frag(Bc0, Mm, half * Dd + k), c0);
      c1 = wmma4(a, b_frag(Bc1, Mm, half * Dd + k), c1);
    }
    __syncthreads();  // everyone done reading before restage
  }
  float bv0 = b1[(2 * w) * 16 + (l & 15)];
  float bv1 = b1[(2 * w + 1) * 16 + (l & 15)];
#pragma unroll
  for (int r = 0; r < 8; ++r) { c0[r] += bv0; c1[r] += bv1; }
  if (w == 0) {
    float th = th1[0];
    rot_tile_cols01(c0, __cosf(th), __sinf(th));
  }
#pragma unroll
  for (int r = 0; r < 8; ++r) { c0[r] = tanhf(c0[r]); c1[r] = tanhf(c1[r]); }
  float* hb = h + (size_t)tm * 16 * Mm;
  store_tile(hb + (2 * w) * 16, Mm, c0);
  store_tile(hb + (2 * w + 1) * 16, Mm, c1);
}

// ---------------------------------------------------------------------------
// K3: routing logits + gumbel softmax (soft) or argmax (hard)
// ---------------------------------------------------------------------------
__global__ __launch_bounds__(256) void k_route(
    const float* __restrict__ h, const float* __restrict__ W2,
    const float* __restrict__ b2, const float* __restrict__ th2,
    const float* __restrict__ gum, float* __restrict__ rw,
    int* __restrict__ psel, int hard) {
  int row = blockIdx.x * 8 + (threadIdx.x >> 5);
  int l = laneid();
  float part[Pp];
#pragma unroll
  for (int p = 0; p < Pp; ++p) part[p] = 0.f;
  for (int m = l; m < Mm; m += 32) {
    float hv = h[(size_t)row * Mm + m];
#pragma unroll
    for (int p = 0; p < Pp; ++p) part[p] += hv * W2[m * Pp + p];
  }
#pragma unroll
  for (int p = 0; p < Pp; ++p)
#pragma unroll
    for (int off = 16; off; off >>= 1) part[p] += __shfl_xor(part[p], off, 32);
  float lg[Pp];
#pragma unroll
  for (int p = 0; p < Pp; ++p) lg[p] = part[p] + b2[p];
  float cth = __cosf(th2[0]), sth = __sinf(th2[0]);
  float l0 = lg[0], l1 = lg[1];
  lg[0] = l0 * cth - l1 * sth;
  lg[1] = l0 * sth + l1 * cth;
#pragma unroll
  for (int p = 0; p < Pp; ++p)
    lg[p] = (lg[p] + gum[(size_t)row * Pp + p]) / TAUv;
  if (hard) {
    int best = 0; float bvv = lg[0];
#pragma unroll
    for (int p = 1; p < Pp; ++p) if (lg[p] > bvv) { bvv = lg[p]; best = p; }
    if (l == 0) psel[row] = best;
  } else {
    float mx = lg[0];
#pragma unroll
    for (int p = 1; p < Pp; ++p) mx = fmaxf(mx, lg[p]);
    float se = 0.f;
#pragma unroll
    for (int p = 0; p < Pp; ++p) se += __expf(lg[p] - mx);
    if (l < Pp) rw[(size_t)row * Pp + l] = __expf(lg[l] - mx) / se;
  }
}

// ---------------------------------------------------------------------------
// K4: twist  tin = xm + rot(col @ W + b, th)   [BS, D]
// A block staged by TDM (or async fallback); wave w owns 8 col tiles.
// ---------------------------------------------------------------------------
__global__ __launch_bounds__(256) void k_twist(
    const float* __restrict__ col, const float* __restrict__ W,
    const float* __restrict__ bias, const float* __restrict__ th,
    const float* __restrict__ xm, float* __restrict__ tin) {
  __shared__ __align__(16) float a_s[16 * APAD];
  int tm = blockIdx.x;
  int w = threadIdx.x >> 5;
  int l = laneid();
  stageA(a_s, col + (size_t)tm * 16 * Dd);
  __syncthreads();
  v8f cb[8];
#pragma unroll
  for (int t = 0; t < 8; ++t) { v8f z = {}; cb[t] = z; }
  for (int k = 0; k < Dd; k += 4) {
    v2f a = a_fragA(a_s, k);
#pragma unroll
    for (int t = 0; t < 8; ++t)
      cb[t] = wmma4(a, b_frag(W + (w * 8 + t) * 16, Dd, k), cb[t]);
  }
  float t0 = th[0];
  float cth = __cosf(t0), sth = __sinf(t0);
  int colx = l & 15, rhi = (l >> 4) << 3;
#pragma unroll
  for (int t = 0; t < 8; ++t) {
    int n0 = (w * 8 + t) * 16;
    float bv = bias[n0 + colx];
#pragma unroll
    for (int r = 0; r < 8; ++r) cb[t][r] += bv;
    if (w == 0 && t == 0) rot_tile_cols01(cb[t], cth, sth);
    const float* xmr = xm + (size_t)tm * 16 * Dd + n0;
#pragma unroll
    for (int r = 0; r < 8; ++r) cb[t][r] += xmr[(size_t)(r + rhi) * Dd + colx];
    store_tile(tin + (size_t)tm * 16 * Dd + n0, Dd, cb[t]);
  }
}

// ---------------------------------------------------------------------------
// K5: fused per-path GEMMs + soft collapse
// ---------------------------------------------------------------------------
__global__ __launch_bounds__(256) void k_path(
    const float* __restrict__ tin, const float* __restrict__ rw,
    const float* __restrict__ Wa, const float* __restrict__ ba,
    const float* __restrict__ tha, const float* __restrict__ Wb,
    const float* __restrict__ bb, const float* __restrict__ thb,
    float* __restrict__ acc) {
  __shared__ __align__(16) float a_s[16 * APAD];  // 65.8 KB
  __shared__ float h_s[16 * HPAD];                // 16.6 KB
  __shared__ float rw_s[16 * Pp];
  int w = threadIdx.x >> 5;
  int l = laneid();
  int tm = blockIdx.x;
  if (threadIdx.x < 16 * Pp)
    rw_s[threadIdx.x] = rw[(size_t)tm * 16 * Pp + threadIdx.x];
  stageA(a_s, tin + (size_t)tm * 16 * Dd);
  v8f accP[8];
#pragma unroll
  for (int t = 0; t < 8; ++t) { v8f z = {}; accP[t] = z; }
  __syncthreads();

  for (int p = 0; p < Pp; ++p) {
    // pull next path's weights toward L2 while we compute this one
    if (p + 1 < Pp) {
      __builtin_prefetch(Wa + (size_t)(p + 1) * Dd * Mm + threadIdx.x * 16, 0, 0);
      __builtin_prefetch(Wb + (size_t)(p + 1) * Mm * Dd + threadIdx.x * 16, 0, 0);
    }
    // ---- GEMM A: h tile cols [2w*16, 2w*16+32) ----
    const float* Ba = Wa + (size_t)p * Dd * Mm;
    v8f c0 = {}, c1 = {};
    for (int k = 0; k < Dd; k += 4) {
      v2f a = a_fragA(a_s, k);
      c0 = wmma4(a, b_frag(Ba + (2 * w) * 16, Mm, k), c0);
      c1 = wmma4(a, b_frag(Ba + (2 * w + 1) * 16, Mm, k), c1);
    }
    float bv0 = ba[p * Mm + (2 * w) * 16 + (l & 15)];
    float bv1 = ba[p * Mm + (2 * w + 1) * 16 + (l & 15)];
#pragma unroll
    for (int r = 0; r < 8; ++r) { c0[r] += bv0; c1[r] += bv1; }
    if (w == 0) {
      float ta = tha[p];
      rot_tile_cols01(c0, __cosf(ta), __sinf(ta));
    }
    store_tile(h_s + (2 * w) * 16, HPAD, c0);
    store_tile(h_s + (2 * w + 1) * 16, HPAD, c1);
    __syncthreads();

    // ---- GEMM B: out tiles cols [w*8*16, (w*8+8)*16) ----
    const float* Bw = Wb + (size_t)p * Mm * Dd;
    v8f cb[8];
#pragma unroll
    for (int t = 0; t < 8; ++t) { v8f z = {}; cb[t] = z; }
    const float* hp = h_s;
    for (int k = 0; k < Mm; k += 4) {
      v2f a = a_frag(hp, HPAD, k);
#pragma unroll
      for (int t = 0; t < 8; ++t)
        cb[t] = wmma4(a, b_frag(Bw + (w * 8 + t) * 16, Dd, k), cb[t]);
    }
    float tb = thb[p];
    float cbt = __cosf(tb), sbt = __sinf(tb);
#pragma unroll
    for (int t = 0; t < 8; ++t) {
      int n0 = (w * 8 + t) * 16;
      float bv = bb[(size_t)p * Dd + n0 + (l & 15)];
#pragma unroll
      for (int r = 0; r < 8; ++r) cb[t][r] += bv;
      if (w == 0 && t == 0) rot_tile_cols01(cb[t], cbt, sbt);
#pragma unroll
      for (int r = 0; r < 8; ++r) {
        int rowl = r + ((l >> 4) << 3);
        accP[t][r] += rw_s[rowl * Pp + p] * cb[t][r];
      }
    }
    __syncthreads();  // h_s consumed; safe to overwrite next p
  }
#pragma unroll
  for (int t = 0; t < 8; ++t)
    store_tile(acc + (size_t)tm * 16 * Dd + (w * 8 + t) * 16, Dd, accP[t]);
}

// ---------------------------------------------------------------------------
// K6: mod-phase (channel%7 group) RMS norm; optional elementwise addend.
// ---------------------------------------------------------------------------
__global__ __launch_bounds__(256) void k_norm(const float* __restrict__ in,
                                              const float* __restrict__ add,
                                              const float* __restrict__ g,
                                              float* __restrict__ out) {
  __shared__ float grp[7];
  int row = blockIdx.x, t = threadIdx.x;
  if (t < 7) grp[t] = 0.f;
  __syncthreads();
  float vals[4];
#pragma unroll
  for (int j = 0; j < 4; ++j) {
    int d = t + j * 256;
    float v = in[(size_t)row * Dd + d];
    if (add) v += add[(size_t)row * Dd + d];
    vals[j] = v;
    atomicAdd(&grp[d % 7], v * v);
  }
  __syncthreads();
#pragma unroll
  for (int j = 0; j < 4; ++j) {
    int d = t + j * 256;
    int gi = d % 7;
    float cnt = (float)((Dd + 6 - gi) / 7);
    out[(size_t)row * Dd + d] =
        vals[j] * rsqrtf(grp[gi] / cnt + EPSv) * g[d];
  }
}

// ---------------------------------------------------------------------------
// K7: final hard gather: out = norm(x + path_out[row, psel[row], :], g)
// ---------------------------------------------------------------------------
__global__ __launch_bounds__(256) void k_final(
    const float* __restrict__ tin, const int* __restrict__ psel,
    const float* __restrict__ Wa, const float* __restrict__ ba,
    const float* __restrict__ tha, const float* __restrict__ Wb,
    const float* __restrict__ bb, const float* __restrict__ thb,
    const float* __restrict__ x, const float* __restrict__ g,
    float* __restrict__ out) {
  __shared__ float h_s[Mm];
  __shared__ float row_s[Dd];
  __shared__ float grp[7];
  int row = blockIdx.x, t = threadIdx.x;
  int p = psel[row];
  const float* tr = tin + (size_t)row * Dd;
  const float* wa = Wa + (size_t)p * Dd * Mm;
  float acc = ba[p * Mm + t];
  for (int d = 0; d < Dd; ++d) acc += tr[d] * wa[(size_t)d * Mm + t];
  h_s[t] = acc;
  if (t < 7) grp[t] = 0.f;
  __syncthreads();
  if (t == 0) {
    float th = tha[p], c = __cosf(th), s = __sinf(th);
    float h0 = h_s[0], h1 = h_s[1];
    h_s[0] = h0 * c - h1 * s;
    h_s[1] = h0 * s + h1 * c;
  }
  __syncthreads();
  const float* wb = Wb + (size_t)p * Mm * Dd;
  float o[4];
#pragma unroll
  for (int j = 0; j < 4; ++j) o[j] = bb[(size_t)p * Dd + 4 * t + j];
  for (int m = 0; m < Mm; ++m) {
    float hv = h_s[m];
    const float4 wv = *(const float4*)(wb + (size_t)m * Dd + 4 * t);
    o[0] += hv * wv.x; o[1] += hv * wv.y; o[2] += hv * wv.z; o[3] += hv * wv.w;
  }
  if (t == 0) {  // channels 0,1 both live in thread 0's chunk
    float th = thb[p], c = __cosf(th), s = __sinf(th);
    float o0 = o[0], o1 = o[1];
    o[0] = o0 * c - o1 * s;
    o[1] = o0 * s + o1 * c;
  }
#pragma unroll
  for (int j = 0; j < 4; ++j) {
    int d = 4 * t + j;
    float v = x[(size_t)row * Dd + d] + o[j];
    row_s[d] = v;
    atomicAdd(&grp[d % 7], v * v);
  }
  __syncthreads();
#pragma unroll
  for (int j = 0; j < 4; ++j) {
    int d = 4 * t + j;
    int gi = d % 7;
    float cnt = (float)((Dd + 6 - gi) / 7);
    out[(size_t)row * Dd + d] = row_s[d] * rsqrtf(grp[gi] / cnt + EPSv) * g[d];
  }
}

// ---------------------------------------------------------------------------
extern "C" void kernel_launch(void* const* d_in, const int* in_sizes, int n_in,
                              void* d_out, int out_size, void* d_ws,
                              size_t ws_size, hipStream_t stream) {
  (void)in_sizes; (void)n_in; (void)out_size; (void)ws_size;
  const float* x       = (const float*)d_in[0];
  const float* gum     = (const float*)d_in[1];
  const float* mixer_w = (const float*)d_in[2];
  const float* obs_W1  = (const float*)d_in[3];
  const float* obs_b1  = (const float*)d_in[4];
  const float* obs_th1 = (const float*)d_in[5];
  const float* obs_W2  = (const float*)d_in[6];
  const float* obs_b2  = (const float*)d_in[7];
  const float* obs_th2 = (const float*)d_in[8];
  const float* path_Wa = (const float*)d_in[9];
  const float* path_ba = (const float*)d_in[10];
  const float* path_tha= (const float*)d_in[11];
  const float* path_Wb = (const float*)d_in[12];
  const float* path_bb = (const float*)d_in[13];
  const float* path_thb= (const float*)d_in[14];
  const float* twist_W = (const float*)d_in[15];
  const float* twist_b = (const float*)d_in[16];
  const float* twist_th= (const float*)d_in[17];
  const float* norm_g  = (const float*)d_in[18];
  float* outp = (float*)d_out;

  const size_t nBSD = (size_t)BS * Dd;  // 8388608
  float* xm   = (float*)d_ws;
  float* colp = xm + nBSD;
  float* accb = colp + nBSD;
  float* tinb = accb + nBSD;
  float* hob  = tinb + nBSD;                 // BS*M
  float* rwb  = hob + (size_t)BS * Mm;       // BS*P
  int*   psb  = (int*)(rwb + (size_t)BS * Pp);

  hipMemsetAsync(colp, 0, nBSD * sizeof(float), stream);

  k_mix<<<(int)(nBSD / 256), 256, 0, stream>>>(x, mixer_w, xm);

  for (int t = 0; t < 3; ++t) {
    k_obs_h<<<512, 256, 0, stream>>>(xm, colp, obs_W1, obs_b1, obs_th1, hob);
    k_route<<<1024, 256, 0, stream>>>(hob, obs_W2, obs_b2, obs_th2,
                                      gum + (size_t)t * BS * Pp, rwb, psb, 0);
    k_twist<<<512, 256, 0, stream>>>(colp, twist_W, twist_b, twist_th, xm,
                                     tinb);
    k_path<<<512, 256, 0, stream>>>(tinb, rwb, path_Wa, path_ba, path_tha,
                                    path_Wb, path_bb, path_thb, accb);
    k_norm<<<BS, 256, 0, stream>>>(accb, nullptr, norm_g, colp);
  }

  // final hard routing
  k_obs_h<<<512, 256, 0, stream>>>(xm, colp, obs_W1, obs_b1, obs_th1, hob);
  k_route<<<1024, 256, 0, stream>>>(hob, obs_W2, obs_b2, obs_th2,
                                    gum + (size_t)3 * BS * Pp, rwb, psb, 1);
  k_final<<<BS, 256, 0, stream>>>(tinb, psb, path_Wa, path_ba, path_tha,
                                  path_Wb, path_bb, path_thb, x, norm_g, outp);
  k_norm<<<BS, 256, 0, stream>>>(colp, nullptr, norm_g, outp + nBSD);
}